// SRU_29746943492442
// MI455X (gfx1250) — compile-verified
//
#include <hip/hip_runtime.h>
#include <stdint.h>

#define EPS 1e-6f

typedef float v4f __attribute__((ext_vector_type(4)));

__device__ __forceinline__ uint32_t lds_lo(const void* p) {
  // generic pointer to __shared__: low 32 bits == LDS byte offset
  return (uint32_t)(uintptr_t)p;
}

// ds_swizzle_b32, group-of-32 mode: src_lane = ((lane & 0x1f) | 0) ^ XOR
template <int XOR>
__device__ __forceinline__ float swz_xor(float v) {
  return __int_as_float(
      __builtin_amdgcn_ds_swizzle(__float_as_int(v), (XOR << 10) | 0x1f));
}

// ---------------------------------------------------------------------------
// Kernel 1 (tiny, latency-optimized): scale[c]=g*g/sum(g), shift[c]=b*g/sum(g),
// thrLogit = logit(mean over b,c of sigmoid(normed[b, 1568, c])).
// One block, 1024 threads = 32 waves (wave w = batch w, spatial row k=1568).
// ---------------------------------------------------------------------------
__global__ __launch_bounds__(1024) void sru_prep(
    const float* __restrict__ x, const float* __restrict__ gamma,
    const float* __restrict__ beta, float* __restrict__ ws) {
  __shared__ float red[32];
  __shared__ float sc_s[256];
  __shared__ float sh_s[256];
  const int tid  = threadIdx.x;
  const int wave = tid >> 5;
  const int lane = tid & 31;

  // ---- sum(gamma): per-wave swizzle butterfly, then 8 partials via LDS ----
  const float g = (tid < 256) ? gamma[tid] : 0.0f;
  float wsum = g;
  wsum += swz_xor<16>(wsum); wsum += swz_xor<8>(wsum);
  wsum += swz_xor<4>(wsum);  wsum += swz_xor<2>(wsum);
  wsum += swz_xor<1>(wsum);
  if (lane == 0) red[wave] = wsum;
  __syncthreads();
  float sg = 0.f;
  #pragma unroll
  for (int w2 = 0; w2 < 8; ++w2) sg += red[w2];   // waves 8..31 hold zeros

  if (tid < 256) {
    const float f  = g / sg;
    const float sc = g * f;
    const float sh = beta[tid] * f;
    sc_s[tid] = sc; sh_s[tid] = sh;
    ws[tid] = sc; ws[256 + tid] = sh;
  }
  __syncthreads();

  // ---- n = sigmoid(normed) at row k=1568 of batch `wave`; global mean ----
  const float* xp = x + ((size_t)wave * 3136 + 1568) * 256 + lane * 8;
  float xv[8];
  *(float4*)(&xv[0]) = *(const float4*)xp;
  *(float4*)(&xv[4]) = *(const float4*)(xp + 4);

  float s = 0.f, ssq = 0.f;
  #pragma unroll
  for (int j = 0; j < 8; ++j) { s += xv[j]; ssq += xv[j] * xv[j]; }
  s   += swz_xor<1>(s);            // group = 16 channels = 2 lanes
  ssq += swz_xor<1>(ssq);
  const float mean = s * 0.0625f;
  const float rinv = rsqrtf(ssq * 0.0625f - mean * mean + EPS);

  float part = 0.f;
  #pragma unroll
  for (int j = 0; j < 8; ++j) {
    const int c = lane * 8 + j;
    const float nrm = (xv[j] - mean) * rinv * sc_s[c] + sh_s[c];
    part += 1.0f / (1.0f + expf(-nrm));
  }
  part += swz_xor<16>(part); part += swz_xor<8>(part);
  part += swz_xor<4>(part);  part += swz_xor<2>(part);
  part += swz_xor<1>(part);
  if (lane == 0) red[wave] = part;
  __syncthreads();
  if (tid == 0) {
    float tot = 0.f;
    #pragma unroll
    for (int w2 = 0; w2 < 32; ++w2) tot += red[w2];
    const float thr = tot * (1.0f / 8192.0f);      // mean over 32*256 values
    ws[512] = logf(thr / (1.0f - thr));            // logit(thr): no sigmoid in pass 2
  }
}

// ---------------------------------------------------------------------------
// Kernel 2 (streaming, ~206MB -> ~8.8us HBM floor at 23.3 TB/s):
// 1 wave per pixel (256 ch), lane l owns channels 8l..8l+7.
// Double-buffered NT async global->LDS pipeline, 8 pixels per wave.
// Mask test hoisted to: (x-mean)*(rinv*sc[j]) > (thrLogit - sh[j]),
// with th2[j] kernel-invariant. u floats cross lanes via ds_swizzle SWAP,16.
// ---------------------------------------------------------------------------
__global__ __launch_bounds__(256) void sru_main(
    const float* __restrict__ x, const float* __restrict__ ws,
    float* __restrict__ out) {
  __shared__ float smem[8 * 2 * 256];   // 8 waves * 2 buffers * 1KB = 16 KB
  const int lane = threadIdx.x & 31;
  const int wave = threadIdx.x >> 5;

  const float thrLogit = ws[512];
  float sc[8], th2[8];
  *(float4*)(&sc[0]) = *(const float4*)(ws + lane * 8);          // RT: stays in L2
  *(float4*)(&sc[4]) = *(const float4*)(ws + lane * 8 + 4);
  {
    float sh[8];
    *(float4*)(&sh[0]) = *(const float4*)(ws + 256 + lane * 8);
    *(float4*)(&sh[4]) = *(const float4*)(ws + 256 + lane * 8 + 4);
    #pragma unroll
    for (int j = 0; j < 8; ++j) th2[j] = thrLogit - sh[j];       // hoisted
  }

  const size_t posBase = ((size_t)blockIdx.x * 8 + wave) * 8;    // 8 pixels/wave
  const uint32_t ldsBase = lds_lo(smem + wave * 512) + (uint32_t)(lane * 16);

  // prologue: async-load pixel 0 into buffer 0 (2 x b128 per lane = 1KB/wave)
  {
    uint64_t ga = (uint64_t)(uintptr_t)(x + posBase * 256) + (uint64_t)(lane * 16);
    uint32_t la = ldsBase;
    asm volatile("global_load_async_to_lds_b128 %0, %1, off th:TH_LOAD_NT"
                 :: "v"(la), "v"(ga) : "memory");
    uint64_t gb = ga + 512; uint32_t lb = la + 512;
    asm volatile("global_load_async_to_lds_b128 %0, %1, off th:TH_LOAD_NT"
                 :: "v"(lb), "v"(gb) : "memory");
  }

  #pragma unroll
  for (int i = 0; i < 8; ++i) {
    if (i < 7) {
      // issue next pixel into the other buffer, then wait for current (<=2 in flight)
      uint64_t ga = (uint64_t)(uintptr_t)(x + (posBase + i + 1) * 256) + (uint64_t)(lane * 16);
      uint32_t la = ldsBase + ((uint32_t)((i + 1) & 1)) * 1024u;
      asm volatile("global_load_async_to_lds_b128 %0, %1, off th:TH_LOAD_NT"
                   :: "v"(la), "v"(ga) : "memory");
      uint64_t gb = ga + 512; uint32_t lb = la + 512;
      asm volatile("global_load_async_to_lds_b128 %0, %1, off th:TH_LOAD_NT"
                   :: "v"(lb), "v"(gb) : "memory");
      asm volatile("s_wait_asynccnt 2" ::: "memory");
    } else {
      asm volatile("s_wait_asynccnt 0" ::: "memory");
    }

    const float* b = smem + wave * 512 + (i & 1) * 256 + lane * 8;
    float xv[8];
    *(float4*)(&xv[0]) = *(const float4*)b;        // ds_load_b128 x2
    *(float4*)(&xv[4]) = *(const float4*)(b + 4);

    float s = 0.f, ssq = 0.f;
    #pragma unroll
    for (int j = 0; j < 8; ++j) { s += xv[j]; ssq += xv[j] * xv[j]; }
    s   += swz_xor<1>(s);                          // group = 16 ch = lane pair
    ssq += swz_xor<1>(ssq);
    const float mean = s * 0.0625f;
    const float rinv = rsqrtf(ssq * 0.0625f - mean * mean + EPS);

    float asc[8];
    #pragma unroll
    for (int j = 0; j < 8; ++j) asc[j] = rinv * sc[j];   // per-pixel setup

    float t[8], u[8];
    #pragma unroll
    for (int j = 0; j < 8; ++j) {
      const bool m = ((xv[j] - mean) * asc[j] > th2[j]); // mask, no sigmoid
      t[j] = m ? xv[j] : 0.0f;
      u[j] = m ? 0.0f : xv[j];
    }

    float o[8];
    #pragma unroll
    for (int j = 0; j < 8; ++j)
      o[j] = t[j] + swz_xor<16>(u[j]);                   // pair c <-> c+128

    float* op = out + (posBase + i) * 256 + lane * 8;
    v4f o0 = { o[0], o[1], o[2], o[3] };
    v4f o1 = { o[4], o[5], o[6], o[7] };
    __builtin_nontemporal_store(o0, (v4f*)op);           // th:TH_STORE_NT
    __builtin_nontemporal_store(o1, (v4f*)(op + 4));
  }
}

extern "C" void kernel_launch(void* const* d_in, const int* in_sizes, int n_in,
                              void* d_out, int out_size, void* d_ws, size_t ws_size,
                              hipStream_t stream) {
  const float* x     = (const float*)d_in[0];
  const float* gamma = (const float*)d_in[1];
  const float* beta  = (const float*)d_in[2];
  float* out = (float*)d_out;
  float* ws  = (float*)d_ws;   // [0,256)=scale, [256,512)=shift, [512]=logit(thr)

  sru_prep<<<1, 1024, 0, stream>>>(x, gamma, beta, ws);
  // 1568 blocks * 8 waves * 8 pixels = 100352 = 32*56*56 pixels exactly
  sru_main<<<1568, 256, 0, stream>>>(x, ws, out);
}